// LuongAttention_6614249636284
// MI455X (gfx1250) — compile-verified
//
#include <hip/hip_runtime.h>
#include <hip/hip_bf16.h>

typedef __attribute__((ext_vector_type(2))) float v2f;
typedef __attribute__((ext_vector_type(4))) float v4f;
typedef __attribute__((ext_vector_type(8))) float v8f;

#define BATCH 8
#define TQ 2048
#define TK 4096
#define DIM 128
#define BQ 16                       // q-rows per workgroup
#define ROWSTRIDE (TK + 4)          // padded LDS row stride (floats): 4100 -> stride%64==4, conflict-free
#define NWAVES 8
#define NTHREADS (NWAVES * 32)

// Fused Luong attention: scores = Q K^T ; P = softmax(scores) ; ctx = P K
// One workgroup handles one (batch, 16-row q tile); the full 16 x 4096 score
// block lives in LDS (262 KB of the 320 KB WGP LDS) so scores never touch HBM.
__global__ __launch_bounds__(NTHREADS, 1)
void luong_attn_fused(const float* __restrict__ Q,
                      const float* __restrict__ K,
                      float* __restrict__ ctx_out,   // [B, TQ, DIM]
                      float* __restrict__ w_out)     // [B, TQ, TK]
{
    extern __shared__ float sS[];          // [BQ][ROWSTRIDE] : scores -> exp(scores - max)
    __shared__ float row_inv[BQ];          // 1 / rowsum(exp)

    const int tid   = threadIdx.x;
    const int wave  = tid >> 5;            // 0..7
    const int lane  = tid & 31;
    const int lhalf = lane >> 4;           // 0 or 1 (half-wave)
    const int l16   = lane & 15;

    const int b     = blockIdx.x / (TQ / BQ);
    const int qbase = (blockIdx.x % (TQ / BQ)) * BQ;

    const float* __restrict__ Qb = Q + ((size_t)b * TQ + qbase) * DIM;
    const float* __restrict__ Kb = K + (size_t)b * TK * DIM;

    // ------------------------------------------------------------------
    // Load Q tile as 32 fp32 WMMA A-fragments kept in registers.
    // A 16x4 layout (ISA 7.12.2): lane L, vgpr v  <->  M = L%16, Kdim = v + 2*(L/16)
    // a_frag[j] covers d in [4j, 4j+3].
    // ------------------------------------------------------------------
    v2f a_frag[DIM / 4];
    {
        const float* p = Qb + (size_t)l16 * DIM + 2 * lhalf;
        #pragma unroll
        for (int j = 0; j < DIM / 4; ++j)
            a_frag[j] = *(const v2f*)(p + 4 * j);
    }

    // ------------------------------------------------------------------
    // Phase 1: S[16 x TK] = Q * K^T using v_wmma_f32_16x16x4_f32.
    // Wave w computes k-tiles nt = w, w+8, ... Two interleaved accumulator
    // chains hide WMMA latency. B frag: B[k][n] = Kmat[n][d0+k].
    // ------------------------------------------------------------------
    for (int nt = wave; nt < TK / 16; nt += NWAVES) {
        const float* Kt = Kb + (size_t)(nt * 16 + l16) * DIM + 2 * lhalf;
        v8f acc0 = {}; v8f acc1 = {};
        #pragma unroll
        for (int j = 0; j < DIM / 4; j += 2) {
            v2f b0 = *(const v2f*)(Kt + 4 * j);
            v2f b1 = *(const v2f*)(Kt + 4 * (j + 1));
            acc0 = __builtin_amdgcn_wmma_f32_16x16x4_f32(false, a_frag[j],     false, b0,
                                                         (short)0, acc0, false, false);
            acc1 = __builtin_amdgcn_wmma_f32_16x16x4_f32(false, a_frag[j + 1], false, b1,
                                                         (short)0, acc1, false, false);
        }
        // D layout: vgpr r, lane L -> M = r + 8*(L/16), N = L%16
        #pragma unroll
        for (int r = 0; r < 8; ++r)
            sS[(size_t)(r + 8 * lhalf) * ROWSTRIDE + nt * 16 + l16] = acc0[r] + acc1[r];
    }
    __syncthreads();

    // ------------------------------------------------------------------
    // Softmax over each 4096-wide row. Row (2*wave + lhalf) handled by one
    // half-wave; butterfly shuffles give every lane the row max / sum.
    // exp(x - m) overwrites the scores in LDS (unnormalized P), final
    // normalized weights stream to HBM with non-temporal float4 stores.
    // ------------------------------------------------------------------
    {
        const int row = 2 * wave + lhalf;
        float* rowp = sS + (size_t)row * ROWSTRIDE;

        float m = -3.402823466e38f;
        for (int j = l16 * 4; j < TK; j += 64) {
            v4f v = *(const v4f*)(rowp + j);
            m = fmaxf(m, fmaxf(fmaxf(v.x, v.y), fmaxf(v.z, v.w)));
        }
        #pragma unroll
        for (int off = 8; off > 0; off >>= 1)
            m = fmaxf(m, __shfl_xor(m, off, 16));

        float s = 0.f;
        for (int j = l16 * 4; j < TK; j += 64) {
            v4f v = *(const v4f*)(rowp + j);
            v.x = __expf(v.x - m); v.y = __expf(v.y - m);
            v.z = __expf(v.z - m); v.w = __expf(v.w - m);
            *(v4f*)(rowp + j) = v;
            s += (v.x + v.y) + (v.z + v.w);
        }
        #pragma unroll
        for (int off = 8; off > 0; off >>= 1)
            s += __shfl_xor(s, off, 16);

        const float inv = 1.0f / s;
        if (l16 == 0) row_inv[row] = inv;

        float* wrow = w_out + ((size_t)b * TQ + qbase + row) * TK;
        for (int j = l16 * 4; j < TK; j += 64) {
            v4f v = *(const v4f*)(rowp + j);
            v.x *= inv; v.y *= inv; v.z *= inv; v.w *= inv;
            __builtin_nontemporal_store(v, (v4f*)(wrow + j));   // written-once: bypass L2 residency
        }
    }
    __syncthreads();

    // ------------------------------------------------------------------
    // Phase 2: ctx[16 x DIM] = P_unnorm[16 x TK] @ K[TK x DIM], scaled by
    // 1/rowsum at the store. Wave w owns d-columns [16w, 16w+15].
    // Four interleaved accumulator chains over the 4096-deep reduction.
    // ------------------------------------------------------------------
    {
        const int d0 = wave * 16;
        const float* Ap = sS + (size_t)l16 * ROWSTRIDE + 2 * lhalf;  // A: P[l16][k + 2*lhalf + v]
        const float* Kp = Kb + d0 + l16;                             // B: K[k + 2*lhalf + v][d0 + l16]
        v8f acc0 = {}, acc1 = {}, acc2 = {}, acc3 = {};

        for (int k0 = 0; k0 < TK; k0 += 16) {
            v2f a0 = *(const v2f*)(Ap + k0);
            v2f a1 = *(const v2f*)(Ap + k0 + 4);
            v2f a2 = *(const v2f*)(Ap + k0 + 8);
            v2f a3 = *(const v2f*)(Ap + k0 + 12);
            v2f b0, b1, b2, b3;
            const float* kp = Kp + (size_t)(k0 + 2 * lhalf) * DIM;
            b0.x = kp[0];            b0.y = kp[DIM];
            b1.x = kp[4 * DIM];      b1.y = kp[5 * DIM];
            b2.x = kp[8 * DIM];      b2.y = kp[9 * DIM];
            b3.x = kp[12 * DIM];     b3.y = kp[13 * DIM];
            acc0 = __builtin_amdgcn_wmma_f32_16x16x4_f32(false, a0, false, b0, (short)0, acc0, false, false);
            acc1 = __builtin_amdgcn_wmma_f32_16x16x4_f32(false, a1, false, b1, (short)0, acc1, false, false);
            acc2 = __builtin_amdgcn_wmma_f32_16x16x4_f32(false, a2, false, b2, (short)0, acc2, false, false);
            acc3 = __builtin_amdgcn_wmma_f32_16x16x4_f32(false, a3, false, b3, (short)0, acc3, false, false);
        }

        #pragma unroll
        for (int r = 0; r < 8; ++r) {
            const int m = r + 8 * lhalf;
            const float v = (acc0[r] + acc1[r]) + (acc2[r] + acc3[r]);
            ctx_out[((size_t)b * TQ + qbase + m) * DIM + d0 + l16] = v * row_inv[m];
        }
    }
}

extern "C" void kernel_launch(void* const* d_in, const int* in_sizes, int n_in,
                              void* d_out, int out_size, void* d_ws, size_t ws_size,
                              hipStream_t stream) {
    (void)in_sizes; (void)n_in; (void)d_ws; (void)ws_size; (void)out_size;

    const float* Q = (const float*)d_in[0];   // decoder_output  [B, TQ, DIM]
    const float* K = (const float*)d_in[1];   // encoder_outputs [B, TK, DIM]

    float* ctx = (float*)d_out;                                   // [B, TQ, DIM]
    float* wts = (float*)d_out + (size_t)BATCH * TQ * DIM;        // [B, TQ, TK]

    const int    grid = BATCH * (TQ / BQ);            // 1024 workgroups
    const size_t smem = (size_t)BQ * ROWSTRIDE * sizeof(float);   // 262400 B < 320 KB WGP LDS

    // Allow >64KB dynamic LDS (WGP supports 320KB). Host-side, capture-safe.
    (void)hipFuncSetAttribute((const void*)luong_attn_fused,
                              hipFuncAttributeMaxDynamicSharedMemorySize, (int)smem);

    luong_attn_fused<<<grid, NTHREADS, smem, stream>>>(Q, K, ctx, wts);
}